// RCR_73942156968390
// MI455X (gfx1250) — compile-verified
//
#include <hip/hip_runtime.h>
#include <math.h>

// ---------------------------------------------------------------------------
// Problem constants (reference: inp shape (32, 4096, 512) fp32 -> out (32,512))
// ---------------------------------------------------------------------------
#define BATCH 32
#define LSEQ  4096
#define DDIM  512
#define BD    (BATCH * DDIM)   // 16384 outputs

// Native ext-vector types (required for __builtin_nontemporal_load and WMMA)
typedef float v4f __attribute__((ext_vector_type(4)));
typedef float v2f __attribute__((ext_vector_type(2)));
typedef float v8f __attribute__((ext_vector_type(8)));

__device__ __forceinline__ v4f vmax4(v4f a, v4f b) {
    v4f r;
    r.x = __builtin_fmaxf(a.x, b.x);
    r.y = __builtin_fmaxf(a.y, b.y);
    r.z = __builtin_fmaxf(a.z, b.z);
    r.w = __builtin_fmaxf(a.w, b.w);
    return r;
}

// ---------------------------------------------------------------------------
// Pass 1: HBM-bound streaming pass.
// Grid: (nseg, BATCH) blocks of 128 threads. Each thread owns 4 consecutive
// D-columns (float4); 128 threads cover the whole 512-wide row (2 KiB,
// perfectly coalesced across 4 waves). A 3-row register window gives the
// sliding maxes with each row loaded once (NT b128 loads: stream > L2, no
// reuse). Per-segment partial sums (s0+s1+s2 combined) go to d_ws:
//   part[seg][b][d]  (fp32, nseg*64KiB total, L2-resident for pass 2).
// ---------------------------------------------------------------------------
__global__ __launch_bounds__(128) void rcr_stream_kernel(
    const float* __restrict__ x, float* __restrict__ part, int segLen) {
    const int d4  = threadIdx.x;       // 0..127 -> columns d4*4 .. d4*4+3
    const int seg = blockIdx.x;
    const int b   = blockIdx.y;
    const int l0  = seg * segLen;

    const v4f* p = (const v4f*)(x + ((size_t)b * LSEQ + l0) * DDIM) + d4;
    const int rowStride = DDIM / 4;    // in float4 units

    const float NEGINF = -__builtin_inff();

    // Preload 2-row window (always in bounds: l0 <= LSEQ - segLen, segLen >= 4)
    v4f cur = __builtin_nontemporal_load(p); p += rowStride;
    v4f n1  = __builtin_nontemporal_load(p); p += rowStride;

    v4f s0 = {0.f, 0.f, 0.f, 0.f};
    v4f s1 = s0, s2 = s0;

    // Rows with a valid l+2 neighbor inside this loop
    const int inb       = LSEQ - (l0 + 2);
    const int mainIters = segLen < inb ? segLen : inb;

    int i = 0;
#pragma unroll 4
    for (; i < mainIters; ++i) {
        v4f n2 = __builtin_nontemporal_load(p); p += rowStride;
        s0 += cur;
        v4f big = vmax4(cur, n1); s1 += big;
        v4f tri = vmax4(big, n2); s2 += tri;
        cur = n1; n1 = n2;
    }
    // Tail (last segment only, <= 2 iterations): clipped windows == -inf pad
    for (; i < segLen; ++i) {
        v4f n2 = {NEGINF, NEGINF, NEGINF, NEGINF};
        s0 += cur;
        v4f big = vmax4(cur, n1); s1 += big;
        v4f tri = vmax4(big, n2); s2 += tri;
        cur = n1; n1 = n2;
    }

    v4f t = s0 + s1 + s2;   // combine the three scales locally (exact fp32)
    v4f* dst = (v4f*)part + ((size_t)seg * BATCH + b) * (DDIM / 4) + d4;
    *dst = t;
}

// ---------------------------------------------------------------------------
// Pass 2: cross-segment reduction via V_WMMA_F32_16X16X4_F32.
//   y[m] = scale * sum_k part[k][m]   for m in [0, 16384)
// A-matrix (16x4): rows = 16 consecutive outputs, cols = 4 segments.
//   ISA layout: lanes 0-15 hold A[M=lane, K=0..1], lanes 16-31 A[M=lane-16, K=2..3].
// B-matrix (4x16) = all ones -> D[i,j] = sum_k A[i,k] (layout-independent).
// K-chained accumulation in the f32 C/D registers; lanes 0/16 spill M=0..7 /
// M=8..15 per the documented C/D VGPR layout. EXEC is all-ones through the
// WMMA loop (uniform bounds, fully-populated grid: 1024 waves * 16 = 16384).
// ---------------------------------------------------------------------------
__global__ __launch_bounds__(128) void rcr_reduce_wmma_kernel(
    const float* __restrict__ part, float* __restrict__ out,
    int nseg, float scale) {
    const int lane = threadIdx.x & 31;
    const int wave = blockIdx.x * (blockDim.x >> 5) + (threadIdx.x >> 5);
    const int wb   = wave * 16;                 // first of 16 outputs for wave
    const int m    = wb + (lane & 15);          // A-row this lane supplies
    const int kOff = (lane >> 4) << 1;          // lanes 16-31 supply K=2,3

    v8f c = {};                                 // f32 accumulator (C/D)
    v2f bOnes = {1.0f, 1.0f};                   // ones B-matrix

    for (int k = 0; k < nseg; k += 4) {         // nseg is a multiple of 4
        v2f a;
        a.x = part[(size_t)(k + kOff)     * BD + m];
        a.y = part[(size_t)(k + kOff + 1) * BD + m];
        // 8 args: (neg_a, A, neg_b, B, c_mod, C, reuse_a, reuse_b)
        c = __builtin_amdgcn_wmma_f32_16x16x4_f32(
            false, a, false, bOnes, (short)0, c, false, false);
    }

    // D layout: lane 0 holds D[0..7, 0]; lane 16 holds D[8..15, 0].
    if ((lane & 15) == 0) {
        const int mo = wb + (lane >> 4) * 8;
        v4f lo, hi;
        lo.x = c[0] * scale; lo.y = c[1] * scale;
        lo.z = c[2] * scale; lo.w = c[3] * scale;
        hi.x = c[4] * scale; hi.y = c[5] * scale;
        hi.z = c[6] * scale; hi.w = c[7] * scale;
        ((v4f*)out)[(mo >> 2) + 0] = lo;
        ((v4f*)out)[(mo >> 2) + 1] = hi;
    }
}

// ---------------------------------------------------------------------------
// Launch: nseg adapts (powers of two dividing 4096) to fit ws_size;
// everything deterministic, all work on `stream`.
// ---------------------------------------------------------------------------
extern "C" void kernel_launch(void* const* d_in, const int* in_sizes, int n_in,
                              void* d_out, int out_size, void* d_ws, size_t ws_size,
                              hipStream_t stream) {
    (void)in_sizes; (void)n_in; (void)out_size;
    const float* x   = (const float*)d_in[0];
    float*       out = (float*)d_out;
    float*       ws  = (float*)d_ws;

    int nseg = 64;                              // 64 * 64 KiB = 4 MiB workspace
    while (nseg > 4 && (size_t)nseg * BD * sizeof(float) > ws_size) nseg >>= 1;
    const int segLen = LSEQ / nseg;

    dim3 g1(nseg, BATCH);                       // up to 2048 blocks / 8192 waves
    rcr_stream_kernel<<<g1, 128, 0, stream>>>(x, ws, segLen);

    const float scale = 1.0f / (3.0f * (float)LSEQ);
    // 16384 outputs / 16 per wave = 1024 waves = 256 blocks of 4 waves
    rcr_reduce_wmma_kernel<<<BD / 16 / 4, 128, 0, stream>>>(ws, out, nseg, scale);
}